// ResourceAllocationUnrolled_40381282517066
// MI455X (gfx1250) — compile-verified
//
#include <hip/hip_runtime.h>
#include <hip/hip_bf16.h>
#include <stdint.h>

#define F_EPS 1e-10f
#define K_ROWS 4096
#define U_COLS 512
#define ROWS_PER_BLOCK 8
#define NUM_LAYERS 20

typedef uint32_t u32x4 __attribute__((ext_vector_type(4)));
typedef int      i32x8 __attribute__((ext_vector_type(8)));
typedef int      i32x4 __attribute__((ext_vector_type(4)));

#if __has_builtin(__builtin_amdgcn_tensor_load_to_lds)
#define HAVE_TDM 1
#else
#define HAVE_TDM 0
#endif

__device__ __forceinline__ float wave_reduce_sum32(float v) {
#pragma unroll
  for (int off = 16; off > 0; off >>= 1)
    v += __shfl_xor(v, off, 32);
  return v;
}

#if HAVE_TDM
// Build a 2D TDM descriptor (D#) and issue tensor_load_to_lds (6-arg form:
// g0[4], g1[8], g2[4], g3[4], extra[8], cpol).
// Tile: rows x cols f32, row stride == cols (contiguous tile of a row-major array).
__device__ __forceinline__ void tdm_load_tile_f32(const float* gsrc,
                                                  uint32_t lds_byte_off,
                                                  uint32_t rows, uint32_t cols) {
  uint64_t ga = (uint64_t)(uintptr_t)gsrc;
  u32x4 g0;
  g0[0] = 1u;                                            // count=1, user mode
  g0[1] = lds_byte_off;                                  // lds_addr
  g0[2] = (uint32_t)ga;                                  // global_addr[31:0]
  g0[3] = (uint32_t)((ga >> 32) & 0x1FFFFFFull)          // global_addr[56:32]
          | (2u << 30);                                  // type=2 ("image")
  i32x8 g1;
  g1[0] = 0x00020000;                                    // data_size=2 (4 bytes)
  g1[1] = (int)(cols << 16);                             // tensor_dim0 lo16
  g1[2] = (int)(rows << 16);                             // tensor_dim0 hi16 | tensor_dim1 lo16
  g1[3] = (int)(cols << 16);                             // tensor_dim1 hi16 | tile_dim0
  g1[4] = (int)rows;                                     // tile_dim1 (tile_dim2=0)
  g1[5] = (int)cols;                                     // tensor_dim0_stride lo32
  g1[6] = 0;                                             // stride0 hi16 | stride1 lo16
  g1[7] = 0;                                             // stride1 hi32
  i32x4 z4 = {0, 0, 0, 0};                               // groups 2/3 unused (2D)
  i32x8 z8 = {0, 0, 0, 0, 0, 0, 0, 0};
  __builtin_amdgcn_tensor_load_to_lds(g0, g1, z4, z4, z8, 0);
}
#endif

__global__ __launch_bounds__(256) void ra_layers_kernel(
    const float* __restrict__ Hg, const float* __restrict__ Pg,
    const float* __restrict__ n0p, const float* __restrict__ step_sizes,
    const float* __restrict__ a_init, float* __restrict__ out) {
  __shared__ float ldsH[ROWS_PER_BLOCK * U_COLS];
  __shared__ float ldsP[ROWS_PER_BLOCK * U_COLS];

  const int wave = threadIdx.x >> 5;
  const int lane = threadIdx.x & 31;
  const int row0 = blockIdx.x * ROWS_PER_BLOCK;
  const int row  = row0 + wave;

#if HAVE_TDM
  if (threadIdx.x == 0) {
    tdm_load_tile_f32(Hg + (size_t)row0 * U_COLS,
                      (uint32_t)(uintptr_t)&ldsH[0], ROWS_PER_BLOCK, U_COLS);
    tdm_load_tile_f32(Pg + (size_t)row0 * U_COLS,
                      (uint32_t)(uintptr_t)&ldsP[0], ROWS_PER_BLOCK, U_COLS);
    __builtin_amdgcn_s_wait_tensorcnt(0);
  }
#else
  for (int i = threadIdx.x; i < ROWS_PER_BLOCK * U_COLS; i += 256) {
    ldsH[i] = Hg[(size_t)row0 * U_COLS + i];
    ldsP[i] = Pg[(size_t)row0 * U_COLS + i];
  }
#endif
  __syncthreads();

  // Persistent per-lane state: 16 columns per lane (cols 128*j + 4*lane + t).
  float Pv[16], Hv[16], PHv[16], sPH[16], a[16];
  const float* aBase = a_init + (size_t)row * U_COLS;
#pragma unroll
  for (int j = 0; j < 4; ++j) {
    const int col = 128 * j + 4 * lane;
    const float4 h4 = *(const float4*)&ldsH[wave * U_COLS + col];
    const float4 p4 = *(const float4*)&ldsP[wave * U_COLS + col];
    const float4 a4 = *(const float4*)&aBase[col];
    const float hv[4] = {h4.x, h4.y, h4.z, h4.w};
    const float pv[4] = {p4.x, p4.y, p4.z, p4.w};
    const float av[4] = {a4.x, a4.y, a4.z, a4.w};
#pragma unroll
    for (int t = 0; t < 4; ++t) {
      const int i = 4 * j + t;
      Hv[i]  = hv[t];
      Pv[i]  = pv[t];
      PHv[i] = hv[t] * pv[t];
      sPH[i] = sqrtf(PHv[i]);
      a[i]   = 1.0f / (1.0f + __expf(-av[t]));   // sigmoid(a_init)
    }
  }

  const float n0v = n0p[0];

  for (int l = 0; l < NUM_LAYERS; ++l) {
    const float lr = step_sizes[l];

    // Pass 1: aPH and row-sum.
    float aPH[16];
    float lsum = 0.0f;
#pragma unroll
    for (int i = 0; i < 16; ++i) {
      aPH[i] = a[i] * PHv[i];
      lsum += aPH[i];
    }
    const float S = wave_reduce_sum32(lsum);

    // Pass 2: elementwise terms + second row-sum (c).
    float gd[16], gt[16];
    float csum = 0.0f;
#pragma unroll
    for (int i = 0; i < 16; ++i) {
      const float interf = n0v + S - aPH[i];
      const float Q = sqrtf(aPH[i] + F_EPS);
      const float w = Q / (interf + F_EPS);
      const float s = 2.0f * w * Q - w * w * interf;
      const float inv1s = 1.0f / (1.0f + s);
      const float t = w * w * inv1s;
      gd[i] = w * sPH[i] * inv1s * rsqrtf(a[i] + F_EPS);
      gt[i] = t * PHv[i];
      csum += t * Pv[i];
    }
    const float c = wave_reduce_sum32(csum);

    // Pass 3: gradient step + clamp to [0,1].
#pragma unroll
    for (int i = 0; i < 16; ++i) {
      const float upd = a[i] + lr * (gd[i] + gt[i] - Hv[i] * c);
      a[i] = fminf(fmaxf(upd, 0.0f), 1.0f);
    }
  }

  float* oBase = out + (size_t)row * U_COLS;
#pragma unroll
  for (int j = 0; j < 4; ++j) {
    const int col = 128 * j + 4 * lane;
    float4 o4;
    o4.x = a[4 * j + 0];
    o4.y = a[4 * j + 1];
    o4.z = a[4 * j + 2];
    o4.w = a[4 * j + 3];
    *(float4*)&oBase[col] = o4;
  }
}

extern "C" void kernel_launch(void* const* d_in, const int* in_sizes, int n_in,
                              void* d_out, int out_size, void* d_ws, size_t ws_size,
                              hipStream_t stream) {
  (void)in_sizes; (void)n_in; (void)d_ws; (void)ws_size; (void)out_size;
  const float* H          = (const float*)d_in[0];
  const float* P          = (const float*)d_in[1];
  const float* n0         = (const float*)d_in[2];
  const float* step_sizes = (const float*)d_in[3];
  const float* a_init     = (const float*)d_in[4];
  // d_in[5] threshold_params: unused by the reference forward pass.
  float* out = (float*)d_out;

  dim3 grid(K_ROWS / ROWS_PER_BLOCK);
  dim3 block(256);
  hipLaunchKernelGGL(ra_layers_kernel, grid, block, 0, stream,
                     H, P, n0, step_sizes, a_init, out);
}